// Aggregator_38543036514382
// MI455X (gfx1250) — compile-verified
//
#include <hip/hip_runtime.h>

#define N_DIM 64
#define N_REL 32

#if __has_builtin(__builtin_amdgcn_global_load_async_to_lds_b128) && \
    __has_builtin(__builtin_amdgcn_s_wait_asynccnt)
#define USE_ASYNC_LDS 1
#else
#define USE_ASYNC_LDS 0
#endif

typedef int v4i __attribute__((ext_vector_type(4)));

// ---------------------------------------------------------------------------
// Zero the accumulator (d_out) and the per-entity counts (d_ws).
// Must run every launch: harness poisons once and never re-poisons.
// ---------------------------------------------------------------------------
__global__ void zero_ws_kernel(float4* __restrict__ out, float* __restrict__ counts,
                               int n_out4, int n_counts) {
    const int stride = gridDim.x * blockDim.x;
    const int tid = blockIdx.x * blockDim.x + threadIdx.x;
    const float4 z = {0.0f, 0.0f, 0.0f, 0.0f};
    for (int i = tid; i < n_out4; i += stride) out[i] = z;
    for (int i = tid; i < n_counts; i += stride) counts[i] = 0.0f;
}

// ---------------------------------------------------------------------------
// Half-wave (16 lanes) per edge; lane slot sl owns elements [4*sl, 4*sl+3].
// Two edges per wave32 per iteration.
//  - relation table (32x64 f32 = 8KB) staged in LDS via async-to-LDS (ASYNCcnt)
//  - software pipeline: next edge's indices loaded early + tail row prefetched
//  - scatter via global_atomic_add_f32 resolved in the L2 atomic units
//    (entity table + accumulator both fit in the 192MB L2)
// ---------------------------------------------------------------------------
__global__ __launch_bounds__(256)
void rgcn_scatter_kernel(const float* __restrict__ emb,
                         const int* __restrict__ head,
                         const int* __restrict__ tail,
                         const int* __restrict__ etype,
                         const float* __restrict__ weight,
                         float* __restrict__ sums,
                         float* __restrict__ counts,
                         int n_edges) {
    __shared__ __align__(16) float w_lds[N_REL * N_DIM];  // 8 KB of 320 KB WGP LDS

#if USE_ASYNC_LDS
    {
        const int n_chunks = (N_REL * N_DIM * (int)sizeof(float)) / 16;  // 512 x b128
        for (int i = threadIdx.x; i < n_chunks; i += blockDim.x) {
            v4i* src = (v4i*)((const char*)weight + (size_t)i * 16);
            v4i* dst = (v4i*)((char*)w_lds + (size_t)i * 16);
            __builtin_amdgcn_global_load_async_to_lds_b128(src, dst, 0, 0);
        }
        __builtin_amdgcn_s_wait_asynccnt(0);
    }
    __syncthreads();
#else
    for (int i = threadIdx.x; i < N_REL * N_DIM; i += blockDim.x)
        w_lds[i] = weight[i];
    __syncthreads();
#endif

    const int lane = threadIdx.x & 31;   // wave32
    const int sub  = lane >> 4;          // which edge of the wave's pair (0/1)
    const int sl   = lane & 15;          // slot within edge: owns 4 floats
    const int wave = (blockIdx.x * blockDim.x + threadIdx.x) >> 5;
    const int stride = ((gridDim.x * blockDim.x) >> 5) * 2;  // edges per sweep

    int e = wave * 2 + sub;
    int h = 0, t = 0, r = 0;
    if (e < n_edges) { h = head[e]; t = tail[e]; r = etype[e]; }

    while (e < n_edges) {
        // ---- stage next iteration's indices + prefetch its tail row ----
        const int en = e + stride;
        int h2 = 0, t2 = 0, r2 = 0;
        if (en < n_edges) {
            h2 = head[en]; t2 = tail[en]; r2 = etype[en];
            __builtin_prefetch(emb + (size_t)t2 * N_DIM + sl * 4, 0, 1);
        }

        // ---- current edge: gather (b128), multiply, scatter ----
        const float4 v = *(const float4*)(emb + (size_t)t * N_DIM + sl * 4);
        const float4 w = *(const float4*)(&w_lds[r * N_DIM + sl * 4]); // ds_load_b128
        float* dst = sums + (size_t)h * N_DIM + sl * 4;
        atomicAdd(dst + 0, v.x * w.x);   // global_atomic_add_f32 (no return)
        atomicAdd(dst + 1, v.y * w.y);
        atomicAdd(dst + 2, v.z * w.z);
        atomicAdd(dst + 3, v.w * w.w);
        if (sl == 0) atomicAdd(counts + h, 1.0f);

        e = en; h = h2; t = t2; r = r2;
    }
}

// ---------------------------------------------------------------------------
// entity_agg = sums / max(counts, 1)   (exact IEEE divide, matches reference)
// float4 per thread: element block i covers entity (i >> 4) since D == 64.
// ---------------------------------------------------------------------------
__global__ void finalize_kernel(float4* __restrict__ out,
                                const float* __restrict__ counts,
                                int total4) {
    const int stride = gridDim.x * blockDim.x;
    for (int i = blockIdx.x * blockDim.x + threadIdx.x; i < total4; i += stride) {
        const float c = fmaxf(counts[i >> 4], 1.0f);
        float4 v = out[i];
        v.x = v.x / c;
        v.y = v.y / c;
        v.z = v.z / c;
        v.w = v.w / c;
        out[i] = v;
    }
}

extern "C" void kernel_launch(void* const* d_in, const int* in_sizes, int n_in,
                              void* d_out, int out_size, void* d_ws, size_t ws_size,
                              hipStream_t stream) {
    const float* emb        = (const float*)d_in[0];   // [N_ENTITIES, 64] f32
    const int*   edge_index = (const int*)d_in[1];     // [2, E] (int32 on device)
    const int*   etype      = (const int*)d_in[2];     // [E]
    const float* weight     = (const float*)d_in[3];   // [32, 64] f32

    const int n_edges    = in_sizes[2];
    const int n_entities = in_sizes[0] / N_DIM;

    float* sums   = (float*)d_out;   // accumulate mean numerator in-place
    float* counts = (float*)d_ws;    // n_entities floats of scratch

    const int* head = edge_index;            // row 0
    const int* tail = edge_index + n_edges;  // row 1

    zero_ws_kernel<<<2048, 256, 0, stream>>>((float4*)sums, counts,
                                             n_entities * N_DIM / 4, n_entities);
    rgcn_scatter_kernel<<<4096, 256, 0, stream>>>(emb, head, tail, etype, weight,
                                                  sums, counts, n_edges);
    finalize_kernel<<<2048, 256, 0, stream>>>((float4*)sums, counts,
                                              n_entities * N_DIM / 4);
}